// Block_36438502540029
// MI455X (gfx1250) — compile-verified
//
#include <hip/hip_runtime.h>

typedef __attribute__((ext_vector_type(2))) float v2f;
typedef __attribute__((ext_vector_type(8))) float v8f;

#define NPIX 784   // 28*28
#define NT   49    // 16-wide N tiles per image (784/16)

enum { EPI_BN_RELU = 0, EPI_BIAS = 1, EPI_BN_RES = 2 };

// fp32 WMMA GEMM: Y[img] = epi(A[MxK] * X[img][KxN]), N = 784.
// Register blocking: one wave computes MT 16-row M-tiles x NTLS 16-col N-tiles
// (A fragments reused across NTLS, B fragments reused across MT).
// f32 16x16x4 fragment layout (ISA 7.12.2): A row-per-lane (lanes0-15 = M),
// half 0 carries K{0,1}, half 1 K{2,3}; B symmetric column-per-lane;
// C/D: VGPR r -> M = base + r + 8*half, N = ntile*16 + (lane&15).
template<int MT, int NTLS, int EPI>
__global__ __launch_bounds__(256)
void gemm_wmma_kernel(const float* __restrict__ A,
                      const float* __restrict__ X,
                      const float* __restrict__ scale,
                      const float* __restrict__ bias,
                      const float* __restrict__ resid,
                      float* __restrict__ Y,
                      int M, int K, int MCH)
{
    constexpr int NPAIRS = (NT + NTLS - 1) / NTLS;   // 25 for NTLS=2

    const int waveInBlk = threadIdx.x >> 5;
    const int lane = threadIdx.x & 31;
    const int half = lane >> 4;
    const int l15  = lane & 15;

    const int wid    = blockIdx.x * 8 + waveInBlk;
    const int npair  = wid % NPAIRS;
    const int mchunk = (wid / NPAIRS) % MCH;
    const int img    = wid / (NPAIRS * MCH);

    // Clamped column index per N-tile (loads always in-range, EXEC all-ones).
    int nc[NTLS];
    #pragma unroll
    for (int u = 0; u < NTLS; ++u) {
        int n = (npair * NTLS + u) * 16 + l15;
        nc[u] = n < NPIX ? n : (NPIX - 1);
    }

    const float* __restrict__ Xi = X + (size_t)img * K * NPIX;

    v8f acc[MT][NTLS] = {};

    const int kh = 2 * half;
    for (int k0 = 0; k0 < K; k0 += 4) {
        if ((k0 & 63) == 0) {
            // pull the X panel ~64 K-rows ahead into cache (global_prefetch_b8)
            int kp = k0 + 64 < K ? k0 + 64 : K - 1;
            __builtin_prefetch(&Xi[(size_t)kp * NPIX + nc[0]], 0, 1);
        }
        v2f bf[NTLS];
        #pragma unroll
        for (int u = 0; u < NTLS; ++u) {
            bf[u].x = Xi[(size_t)(k0 + kh) * NPIX + nc[u]];
            bf[u].y = Xi[(size_t)(k0 + kh + 1) * NPIX + nc[u]];
        }
        #pragma unroll
        for (int t = 0; t < MT; ++t) {
            int m  = (mchunk * MT + t) * 16 + l15;
            int mc = m < M ? m : (M - 1);        // clamp (pad rows; stores guarded)
            v2f af;
            af.x = A[(size_t)mc * K + k0 + kh];
            af.y = A[(size_t)mc * K + k0 + kh + 1];
            #pragma unroll
            for (int u = 0; u < NTLS; ++u)
                acc[t][u] = __builtin_amdgcn_wmma_f32_16x16x4_f32(
                    false, af, false, bf[u], (short)0, acc[t][u], false, false);
        }
    }

    float* __restrict__ Yi = Y + (size_t)img * M * NPIX;
    const float* __restrict__ Ri = resid + (size_t)img * M * NPIX;
    #pragma unroll
    for (int u = 0; u < NTLS; ++u) {
        if ((npair * NTLS + u) < NT) {           // wave-uniform tile guard
            const int n = (npair * NTLS + u) * 16 + l15;
            #pragma unroll
            for (int t = 0; t < MT; ++t) {
                #pragma unroll
                for (int r = 0; r < 8; ++r) {
                    int m = (mchunk * MT + t) * 16 + r + half * 8;
                    if (m < M) {
                        float v = acc[t][u][r];
                        if constexpr (EPI == EPI_BN_RELU) {
                            v = v * scale[m] + bias[m];
                            v = v > 0.f ? v : 0.f;
                        } else if constexpr (EPI == EPI_BIAS) {
                            v = v + bias[m];
                        } else {
                            v = v * scale[m] + bias[m] + Ri[(size_t)m * NPIX + n];
                        }
                        Yi[(size_t)m * NPIX + n] = v;
                    }
                }
            }
        }
    }
}

// Involution apply: out[b,c,h,w] = sum_{7x7} kern[b, c/16, k, h, w] *
// x1[b, c, h+di, w+dj], zero-padded. Per-pixel kernels -> VALU; bandwidth
// bound (kern 19.7MB + x1 6.4MB + out 6.4MB, L2-resident on 192MB L2).
__global__ __launch_bounds__(256)
void involution_kernel(const float* __restrict__ x1,
                       const float* __restrict__ kern,
                       float* __restrict__ y)
{
    int idx = blockIdx.x * 256 + threadIdx.x;   // 16*128*784 total
    int w = idx % 28;
    int h = (idx / 28) % 28;
    int c = (idx / 784) % 128;
    int b = idx / (784 * 128);
    int g = c >> 4;

    const float* __restrict__ kp = kern + (size_t)(b * 8 + g) * 49 * NPIX + h * 28 + w;
    // base pointer at the (h-3, w-3) corner: tap offsets become immediates
    const float* __restrict__ xb = x1 + (size_t)(b * 128 + c) * NPIX + (h - 3) * 28 + (w - 3);

    float s = 0.f;
    #pragma unroll
    for (int i = 0; i < 7; ++i) {
        int hh = h + i - 3;
        bool hok = (hh >= 0) & (hh < 28);
        #pragma unroll
        for (int j = 0; j < 7; ++j) {
            int ww = w + j - 3;
            bool ok = hok & (ww >= 0) & (ww < 28);
            if (ok)
                s += kp[(i * 7 + j) * NPIX] * xb[i * 28 + j];
        }
    }
    y[idx] = s;
}

extern "C" void kernel_launch(void* const* d_in, const int* in_sizes, int n_in,
                              void* d_out, int out_size, void* d_ws, size_t ws_size,
                              hipStream_t stream)
{
    (void)in_sizes; (void)n_in; (void)out_size; (void)ws_size;
    const float* x  = (const float*)d_in[0];   // [16,512,28,28]
    const float* W1 = (const float*)d_in[1];   // [128,512]
    const float* s1 = (const float*)d_in[2];
    const float* b1 = (const float*)d_in[3];
    const float* Wr = (const float*)d_in[4];   // [32,128]
    const float* sr = (const float*)d_in[5];
    const float* br = (const float*)d_in[6];
    const float* Ws = (const float*)d_in[7];   // [392,32]
    const float* bs = (const float*)d_in[8];
    const float* W3 = (const float*)d_in[9];   // [512,128]
    const float* s3 = (const float*)d_in[10];
    const float* b3 = (const float*)d_in[11];
    float* out = (float*)d_out;

    float* ws   = (float*)d_ws;
    float* out1 = ws;                                   // 16*128*784
    float* wred = out1 + (size_t)16 * 128 * NPIX;       // 16*32*784
    float* kern = wred + (size_t)16 * 32 * NPIX;        // 16*392*784
    float* out2 = kern + (size_t)16 * 392 * NPIX;       // 16*128*784

    // conv1: relu(BN(W1 . x)); M=128, K=512; 16 imgs * 25 pairs = 400 waves
    gemm_wmma_kernel<8, 2, EPI_BN_RELU><<<50, 256, 0, stream>>>(
        W1, x, s1, b1, nullptr, out1, 128, 512, 1);

    // involution reduce: relu(BN(Wr . out1)); M=32, K=128
    gemm_wmma_kernel<2, 2, EPI_BN_RELU><<<50, 256, 0, stream>>>(
        Wr, out1, sr, br, nullptr, wred, 32, 128, 1);

    // involution span: Ws . wred + bs; M=392 (padded to 400), K=32; MCH=5
    gemm_wmma_kernel<5, 2, EPI_BIAS><<<250, 256, 0, stream>>>(
        Ws, wred, nullptr, bs, nullptr, kern, 392, 32, 5);

    // involution apply (per-pixel 7x7 kernels)
    involution_kernel<<<(16 * 128 * NPIX) / 256, 256, 0, stream>>>(out1, kern, out2);

    // conv3 + BN + residual: M=512, K=128; MCH=4 -> 1600 waves -> 200 blocks
    gemm_wmma_kernel<8, 2, EPI_BN_RES><<<200, 256, 0, stream>>>(
        W3, out2, s3, b3, x, out, 512, 128, 4);
}